// Baseline_node2vec_27101243638197
// MI455X (gfx1250) — compile-verified
//
#include <hip/hip_runtime.h>

typedef __attribute__((ext_vector_type(2))) float v2f;
typedef __attribute__((ext_vector_type(8))) float v8f;

#define TILE_M 16
#define KDIM 256              // 2 * EMB_DIM
#define HID 256
#define NWAVES 16             // one 16-wide hidden-column tile per wave
#define THREADS (NWAVES * 32)
#define TILES_PER_WG 64
#define PAIRS_PER_WG (TILES_PER_WG * TILE_M)

// s_wait_asynccnt: builtin if the toolchain has it, else raw SOPP.
#if defined(__has_builtin) && __has_builtin(__builtin_amdgcn_s_wait_asynccnt)
  #define WAIT_ASYNCCNT(n) __builtin_amdgcn_s_wait_asynccnt(n)
#else
  #define WAIT_ASYNCCNT(n) asm volatile("s_wait_asynccnt %0" :: "n"(n) : "memory")
#endif

typedef __attribute__((address_space(3))) float lds_f32;

// 32-bit LDS byte offset of a generic pointer into __shared__ memory.
__device__ __forceinline__ unsigned lds_off(void* p) {
    return (unsigned)(unsigned long long)(lds_f32*)p;   // addrspacecast + ptrtoint
}

// Async 16B gather chunk: memory -> LDS, no VGPR round trip, tracked by ASYNCcnt.
__device__ __forceinline__ void async_copy_b128(const float* gsrc, float* ldst) {
    unsigned d = lds_off(ldst);
    unsigned long long a = (unsigned long long)gsrc;
    asm volatile("global_load_async_to_lds_b128 %0, %1, off"
                 :: "v"(d), "v"(a) : "memory");
}

__launch_bounds__(THREADS)
__global__ void node2vec_mlp_wmma(const float* __restrict__ embs,
                                  const long long* __restrict__ idx,
                                  const float* __restrict__ W1,
                                  const float* __restrict__ b1,
                                  const float* __restrict__ W2,
                                  const float* __restrict__ b2g,
                                  float* __restrict__ out,
                                  int E)
{
    __shared__ float xbuf[2][TILE_M * KDIM];   // 2 x 16 KB: gathered input tiles
    __shared__ float hbuf[2][TILE_M * HID];    // 2 x 16 KB: relu(h) tiles

    const int tid  = threadIdx.x;
    const int lane = tid & 31;
    const int wave = tid >> 5;              // 0..15 -> hidden column tile
    const int col  = wave * 16 + (lane & 15);
    const int koff = (lane >> 4) * 2;       // lanes 0-15: K=0,1 ; lanes 16-31: K=2,3
    const int arow = lane & 15;

    // ---- Preload this wave's W1 slice as 64 B-fragments (registers) ----
    // B layout (4x16, KxN): VGPR0 = rows K=0 (lanes 0-15), K=2 (lanes 16-31);
    //                       VGPR1 = rows K=1, K=3.
    v2f w[64];
    #pragma unroll
    for (int s = 0; s < 64; ++s) {
        const float* p = W1 + (4 * s + koff) * HID + col;
        w[s].x = p[0];
        w[s].y = p[HID];
    }
    const float bias1 = b1[col];

    const long long base = (long long)blockIdx.x * PAIRS_PER_WG;
    int ntiles = 0;
    if (base < E) {
        long long rem = ((long long)E - base + (TILE_M - 1)) / TILE_M;
        ntiles = rem < TILES_PER_WG ? (int)rem : TILES_PER_WG;
    }
    if (ntiles <= 0) return;

    // Issue the async gather of one 16x256 tile (2 x b128 chunks per thread).
    auto stage = [&](int t, int buf) {
        const long long tile = base + (long long)t * TILE_M;
        #pragma unroll
        for (int cc = 0; cc < 2; ++cc) {
            int chunk = tid + cc * THREADS;             // 0..1023
            int r = chunk >> 6;                         // row 0..15
            int q = chunk & 63;                         // 16B chunk in row
            long long p  = tile + r;
            long long pi = (p < E) ? p : 0;
            long long node = idx[2 * pi + (q >> 5)];    // int64 indices
            const float* src = embs + node * 128 + (q & 31) * 4;
            async_copy_b128(src, &xbuf[buf][r * KDIM + q * 4]);
        }
    };

    stage(0, 0);                            // prologue: tile 0 in flight

    for (int t = 0; t < ntiles; ++t) {
        const int cur = t & 1;

        // Pipeline: put tile t+1 in flight, then wait only for tile t
        // (async loads complete in order; 2 outstanding = next tile only).
        if (t + 1 < ntiles) {
            stage(t + 1, (t + 1) & 1);
            WAIT_ASYNCCNT(2);
        } else {
            WAIT_ASYNCCNT(0);
        }
        __syncthreads();                    // barrier 1: xbuf[cur] visible to all waves

        // ---- GEMM1: (16x256) @ (256x16) via 64 x wmma_f32_16x16x4_f32 ----
        v8f acc = {0.f, 0.f, 0.f, 0.f, 0.f, 0.f, 0.f, 0.f};
        #pragma unroll
        for (int s = 0; s < 64; ++s) {
            // A layout (16x4): lane L(<16): {x[L][4s], x[L][4s+1]};
            //                  lane L+16 : {x[L][4s+2], x[L][4s+3]}
            v2f a = *(const v2f*)(&xbuf[cur][arow * KDIM + 4 * s + koff]);
            acc = __builtin_amdgcn_wmma_f32_16x16x4_f32(
                      false, a, false, w[s], (short)0, acc, false, false);
        }

        // ---- bias + ReLU, spill h tile to LDS (C layout: VGPR v -> M=v / v+8) ----
        const int rbase = (lane >> 4) * 8;
        #pragma unroll
        for (int v = 0; v < 8; ++v) {
            float h = acc[v] + bias1;
            h = h > 0.f ? h : 0.f;
            hbuf[cur][(rbase + v) * HID + col] = h;
        }
        __syncthreads();                    // barrier 2: hbuf[cur] ready; xbuf reads done

        // ---- GEMM2: (16x256) @ (256x2), wave 0, lane = 2*row + col ----
        // (hbuf is double-buffered: next overwrite of hbuf[cur] is 2 barriers away)
        if (wave == 0) {
            int m = lane >> 1;
            int c = lane & 1;
            float sum = b2g[c];
            #pragma unroll 8
            for (int j = 0; j < HID; ++j)
                sum += hbuf[cur][m * HID + j] * W2[j * 2 + c];
            long long p = base + (long long)t * TILE_M + m;
            if (p < E) out[p * 2 + c] = sum;
        }
    }
}

extern "C" void kernel_launch(void* const* d_in, const int* in_sizes, int n_in,
                              void* d_out, int out_size, void* d_ws, size_t ws_size,
                              hipStream_t stream) {
    const float*     embs = (const float*)d_in[0];
    const long long* idx  = (const long long*)d_in[1];
    const float*     W1   = (const float*)d_in[2];
    const float*     b1   = (const float*)d_in[3];
    const float*     W2   = (const float*)d_in[4];
    const float*     b2   = (const float*)d_in[5];
    float*           out  = (float*)d_out;

    int E = in_sizes[1] / 2;                         // node_indices is [E, 2]
    int blocks = (E + PAIRS_PER_WG - 1) / PAIRS_PER_WG;

    node2vec_mlp_wmma<<<blocks, THREADS, 0, stream>>>(embs, idx, W1, b1, W2, b2, out, E);
}